// MagnitudeMaxPool2Dt_64991445123084
// MI455X (gfx1250) — compile-verified
//
#include <hip/hip_runtime.h>
#include <stdint.h>

// MagnitudeMaxPool over (2,2,2) windows of complex (re,im) f32 data.
// Input : x_re, x_im  shape (B=2, T=16, H=256, W=256, C=32) f32
// Output: (B, T/2, H/2, W/2, C, 2) f32, last axis = (re, im) of the
//         window element with max |z|^2 (first-max tie-break, dw fastest).
//
// Pure streaming kernel: ~604 MB traffic -> ~26 us at 23.3 TB/s.
// gfx1250 path: async global->LDS staging (ASYNCcnt) + NT stores.

typedef __attribute__((ext_vector_type(2))) float v2f;
typedef __attribute__((ext_vector_type(4))) float v4f;
typedef __attribute__((ext_vector_type(4))) int   v4i;

#define B_   2
#define T_   16
#define H_   256
#define W_   256
#define C_   32
#define TP_  8     // T/2
#define HP_  128   // H/2
#define WP_  128   // W/2
#define TILE_W 32  // output pixels (w') per block
#define THREADS 256

#if defined(__gfx1250__) && __has_builtin(__builtin_amdgcn_global_load_async_to_lds_b128)
#define HAVE_ASYNC 1
#else
#define HAVE_ASYNC 0
#endif

#if HAVE_ASYNC
// Builtin signature (from hipcc diagnostic): param0 = v4i in global
// address space (printed as "int __vector(4) __device__ *"), param1 = v4i in
// LDS address space, then two integer immediates (offset, cpol).
typedef __attribute__((address_space(1))) v4i GV4;
typedef __attribute__((address_space(3))) v4i LV4;
#endif

__global__ __launch_bounds__(THREADS)
void MagnitudeMaxPool2Dt_kernel(const float* __restrict__ xre,
                                const float* __restrict__ xim,
                                float* __restrict__ out)
{
    // 64 KiB staging: [arr(re=0,im=1)][dt][dh][w_local 0..63][c 0..31]
    __shared__ float smem[16384];

    const int tid = threadIdx.x;
    const int bx  = blockIdx.x;
    const int wt  = bx & 3;                // which 32-wide w' tile (WP_/TILE_W = 4)
    const int hp  = (bx >> 2) & (HP_ - 1); // output h'
    const int tp  = (bx >> 9) & (TP_ - 1); // output t'
    const int b   = bx >> 12;              // batch

    // ---- Stage 64 KiB of inputs into LDS ------------------------------
    // 4096 x 16B transfers total; 16 per thread. For each k all 256
    // threads cover exactly half of one contiguous 8 KiB segment, so the
    // segment base is scalar (SGPR) work.
#pragma unroll
    for (int k = 0; k < 16; ++k) {
        const int arr  = k >> 3;          // 0 = re, 1 = im
        const int pair = (k >> 1) & 3;    // (dt,dh)
        const int dt   = pair >> 1;
        const int dh   = pair & 1;
        const float* srcb = arr ? xim : xre;
        const size_t gbase =
            ((((size_t)b * T_ + (size_t)(2 * tp + dt)) * H_ +
              (size_t)(2 * hp + dh)) * W_ + (size_t)(64 * wt)) * C_;
        const float* src = srcb + gbase + (size_t)((k & 1) * 1024 + tid * 4);
        float* dst = &smem[k * 1024 + tid * 4];
#if HAVE_ASYNC
        __builtin_amdgcn_global_load_async_to_lds_b128((GV4*)src, (LV4*)dst,
                                                       /*offset=*/0, /*cpol=*/0);
#else
        *(v4f*)dst = *(const v4f*)src;
#endif
    }
#if HAVE_ASYNC
#if __has_builtin(__builtin_amdgcn_s_wait_asynccnt)
    __builtin_amdgcn_s_wait_asynccnt(0);
#else
    asm volatile("s_wait_asynccnt 0x0" ::: "memory");
#endif
#endif
    __syncthreads();

    // ---- Compute: 32 pixels x 32 channels = 1024 outputs, 4/thread ----
    const int c  = tid & 31;         // lane = channel -> conflict-free LDS,
    const int p0 = (tid >> 5) * 4;   // coalesced 256B stores per pixel

    const size_t obase =
        (((size_t)b * TP_ + tp) * HP_ + hp) * WP_ + (size_t)(wt * TILE_W);

#pragma unroll
    for (int j = 0; j < 4; ++j) {
        const int p = p0 + j;        // local w' in [0,32)
        float bestR = 0.f, bestI = 0.f, bestM = 0.f;
        // window order: q = dt*4 + dh*2 + dw (dw fastest) == jnp.argmax order
#pragma unroll
        for (int q = 0; q < 8; ++q) {
            const int dt = q >> 2;
            const int dh = (q >> 1) & 1;
            const int dw = q & 1;
            const int wl = 2 * p + dw;
            const int o  = ((dt * 2 + dh) * 64 + wl) * 32 + c;
            const float r = smem[o];
            const float i = smem[8192 + o];
            const float m = r * r + i * i;
            if (q == 0) {
                bestR = r; bestI = i; bestM = m;
            } else if (m > bestM) {   // strict > keeps first max
                bestM = m; bestR = r; bestI = i;
            }
        }
        v2f v; v.x = bestR; v.y = bestI;
        v2f* op = (v2f*)(out + ((obase + (size_t)p) * C_ + (size_t)c) * 2);
        __builtin_nontemporal_store(v, op);  // streamed once; keep L2 clean
    }
}

extern "C" void kernel_launch(void* const* d_in, const int* in_sizes, int n_in,
                              void* d_out, int out_size, void* d_ws, size_t ws_size,
                              hipStream_t stream) {
    const float* xre = (const float*)d_in[0];
    const float* xim = (const float*)d_in[1];
    float* out = (float*)d_out;

    // blocks = B * TP * HP * (WP/TILE_W) = 2*8*128*4 = 8192
    dim3 grid(B_ * TP_ * HP_ * (WP_ / TILE_W));
    dim3 block(THREADS);
    hipLaunchKernelGGL(MagnitudeMaxPool2Dt_kernel, grid, block, 0, stream,
                       xre, xim, out);
}